// DeformConv2d_83854941487845
// MI455X (gfx1250) — compile-verified
//
#include <hip/hip_runtime.h>
#include <math.h>

typedef __attribute__((ext_vector_type(2))) float v2f;
typedef __attribute__((ext_vector_type(8))) float v8f;

#define BATCH 4
#define CIN   64
#define HH    128
#define WW    128
#define OUTC  64
#define NPT   9
#define KDIM  (CIN * NPT)      // 576
#define KG    (KDIM / 4)       // 144 wmma steps (16x16x4 f32)
#define HWSZ  (HH * WW)        // 16384
#define BHW   (BATCH * HWSZ)   // 65536
#define LDA   (KDIM + 2)       // 578: LDS stride, 578%64==2 -> conflict-free
#define OMSTR 28               // offmod row stride (27 channels used)

// d_ws layout (in floats):
#define WPM_OFF    0                       // 2*KG*64 = 18432 packed p/m weights
#define WMAIN_OFF  (2 * KG * 64)           // 4*KG*64 = 36864 packed main weights
#define OFFMOD_OFF (WPM_OFF + 2*KG*64 + 4*KG*64)  // BHW*28 offmod buffer

// ---------------------------------------------------------------------------
// Pack weights into WMMA B-fragment order.
// Load in GEMM: lane L reads float2 at [(slice*KG+kg)*32 + L]*2:
//   L<16 : elems = B[4kg+0][n], B[4kg+1][n]   (n = L)
//   L>=16: elems = B[4kg+2][n], B[4kg+3][n]   (n = L-16)
// K index k = c*9 + tap  (tap = 3x3 position / sample point)
// ---------------------------------------------------------------------------
__global__ void pack_weights(const float* __restrict__ w_p,
                             const float* __restrict__ w_m,
                             const float* __restrict__ w_conv,
                             float* __restrict__ ws) {
  int e = blockIdx.x * 256 + threadIdx.x;
  const int NPM = 2 * KG * 64;   // offset+mod pack elements
  const int NMN = 4 * KG * 64;   // main pack elements
  if (e < NPM) {
    int slice = e / (KG * 64);
    int r     = e % (KG * 64);
    int kg = r / 64, t = r % 64;
    int lane = t >> 1, j2 = t & 1;
    int k = 4 * kg + ((lane < 16) ? j2 : 2 + j2);
    int c = k / 9, tap = k % 9;
    int n = (lane & 15) + 16 * slice;       // merged out-channel (27 used)
    float v = 0.f;
    if (n < 18)       v = w_p[(n * CIN + c) * 9 + tap];
    else if (n < 27)  v = w_m[((n - 18) * CIN + c) * 9 + tap];
    ws[WPM_OFF + e] = v;
  } else if (e < NPM + NMN) {
    int e2 = e - NPM;
    int slice = e2 / (KG * 64);
    int r     = e2 % (KG * 64);
    int kg = r / 64, t = r % 64;
    int lane = t >> 1, j2 = t & 1;
    int k = 4 * kg + ((lane < 16) ? j2 : 2 + j2);
    int c = k / 9, tap = k % 9;
    int o = (lane & 15) + 16 * slice;       // main conv out-channel
    ws[WMAIN_OFF + e2] = w_conv[(o * CIN + c) * 9 + tap];
  }
}

// ---------------------------------------------------------------------------
// Kernel 1: merged offset+modulation 3x3 conv as implicit GEMM via
// V_WMMA_F32_16X16X4_F32. One block = 16 pixels of one image row.
// ---------------------------------------------------------------------------
__global__ __launch_bounds__(128)
void offmod_conv(const float* __restrict__ x,
                 const float* __restrict__ b_p,
                 const float* __restrict__ b_m,
                 const float* __restrict__ ws_base,
                 float* __restrict__ offmod) {
  __shared__ float As[16 * LDA];
  int base = blockIdx.x * 16;            // flat pixel base; tile is intra-row
  int b  = base / HWSZ;
  int hw = base % HWSZ;
  int h = hw / WW, w0 = hw % WW;
  int tid = threadIdx.x;

  // Phase 1: zero-padded 3x3xC patches -> LDS A tile (16 x 576)
  for (int e = tid; e < 16 * KDIM; e += 128) {
    int k = e >> 4;                      // e = k*16 + pix: lanes span pixels
    int pix = e & 15;
    int c = k / 9, tap = k % 9;
    int hh = h + tap / 3 - 1;
    int ww = w0 + pix + tap % 3 - 1;
    float v = 0.f;
    if ((unsigned)hh < (unsigned)HH && (unsigned)ww < (unsigned)WW)
      v = x[((size_t)(b * CIN + c) * HH + hh) * WW + ww];
    As[pix * LDA + k] = v;
  }
  __syncthreads();

  int wv = tid >> 5, lane = tid & 31;    // wave32
  if (wv < 2) {                          // 2 N-slices cover 27(->32) channels
    const float* wp = ws_base + WPM_OFF + (size_t)(wv * KG) * 64;
    v8f acc = {0.f, 0.f, 0.f, 0.f, 0.f, 0.f, 0.f, 0.f};
    int mrow = lane & 15;
    int koff = (lane < 16) ? 0 : 2;      // A frag: lanes 16-31 hold K+2,K+3
    for (int kg = 0; kg < KG; ++kg) {
      v2f a  = *(const v2f*)&As[mrow * LDA + 4 * kg + koff];
      v2f bf = *(const v2f*)&wp[(size_t)kg * 64 + lane * 2];
      acc = __builtin_amdgcn_wmma_f32_16x16x4_f32(
          false, a, false, bf, (short)0, acc, false, false);
    }
    int ch = (lane & 15) + 16 * wv;
    if (ch < 27) {
      float bias = (ch < 18) ? b_p[ch] : b_m[ch - 18];
      #pragma unroll
      for (int r = 0; r < 8; ++r) {      // D frag: VGPR r -> M=r (+8 hi half)
        int m = r + ((lane < 16) ? 0 : 8);
        float v = acc[r] + bias;
        if (ch >= 18) v = 1.f / (1.f + __expf(-v));   // sigmoid(mod)
        offmod[(size_t)(base + m) * OMSTR + ch] = v;
      }
    }
  }
}

// ---------------------------------------------------------------------------
// Kernel 2: fused bilinear sampling + main GEMM (16 pixels x 64 outputs).
// x_off never touches global memory: gathered straight into the LDS A tile.
// ---------------------------------------------------------------------------
__global__ __launch_bounds__(128)
void deform_gemm(const float* __restrict__ x,
                 const float* __restrict__ ws_base,
                 const float* __restrict__ offmod,
                 float* __restrict__ out) {
  __shared__ float As[16 * LDA];
  __shared__ int   Sidx[16 * NPT][4];
  __shared__ float Sg[16 * NPT][4];
  int base = blockIdx.x * 16;
  int b  = base / HWSZ;
  int hw = base % HWSZ;
  int h = hw / WW, w0 = hw % WW;
  int tid = threadIdx.x;

  // Phase A: per (pixel, sample-point) geometry (reference algebra, padded img)
  for (int s = tid; s < 16 * NPT; s += 128) {
    int pix = s / NPT, n = s % NPT;
    const float* om = offmod + (size_t)(base + pix) * OMSTR;
    float offx = om[n], offy = om[NPT + n], mod = om[18 + n];
    float px_ = offx + (float)(n / 3 - 1) + (float)(h + 1);
    float py_ = offy + (float)(n % 3 - 1) + (float)(w0 + pix + 1);
    int fx = (int)floorf(px_), fy = (int)floorf(py_);
    int qltx = min(max(fx, 0), HH + 1);
    int qlty = min(max(fy, 0), WW + 1);
    int qrbx = min(max(fx + 1, 0), HH + 1);
    int qrby = min(max(fy + 1, 0), WW + 1);
    float pxc = fminf(fmaxf(px_, 0.f), (float)(HH + 1));
    float pyc = fminf(fmaxf(py_, 0.f), (float)(WW + 1));
    float glt = (1.f + ((float)qltx - pxc)) * (1.f + ((float)qlty - pyc));
    float grb = (1.f - ((float)qrbx - pxc)) * (1.f - ((float)qrby - pyc));
    float glb = (1.f + ((float)qltx - pxc)) * (1.f - ((float)qrby - pyc));
    float grt = (1.f - ((float)qrbx - pxc)) * (1.f + ((float)qlty - pyc));
    auto idx = [](int qx, int qy) -> int {   // padded coord -> x index or -1
      return (qx >= 1 && qx <= HH && qy >= 1 && qy <= WW)
                 ? (qx - 1) * WW + (qy - 1) : -1;
    };
    Sidx[s][0] = idx(qltx, qlty);  Sg[s][0] = glt * mod;
    Sidx[s][1] = idx(qrbx, qrby);  Sg[s][1] = grb * mod;
    Sidx[s][2] = idx(qltx, qrby);  Sg[s][2] = glb * mod;
    Sidx[s][3] = idx(qrbx, qlty);  Sg[s][3] = grt * mod;
  }
  __syncthreads();

  // Phase B: bilinear gathers -> A tile (16 x 576), modulation folded in
  for (int e = tid; e < 16 * KDIM; e += 128) {
    int k = e >> 4;                        // lanes span pixels -> local gathers
    int pix = e & 15;
    int c = k / 9, n = k % 9;
    int s = pix * NPT + n;
    const float* xc = x + (size_t)(b * CIN + c) * HWSZ;
    float v = 0.f;
    #pragma unroll
    for (int j = 0; j < 4; ++j) {
      int id = Sidx[s][j];
      if (id >= 0) v += Sg[s][j] * xc[id];
    }
    As[pix * LDA + k] = v;
  }
  __syncthreads();

  // Phase C: 4 waves x (16 pixels x 16 outputs), 144 f32 WMMA steps each
  int wv = tid >> 5, lane = tid & 31;
  const float* wmain = ws_base + WMAIN_OFF + (size_t)(wv * KG) * 64;
  v8f acc = {0.f, 0.f, 0.f, 0.f, 0.f, 0.f, 0.f, 0.f};
  int mrow = lane & 15;
  int koff = (lane < 16) ? 0 : 2;
  for (int kg = 0; kg < KG; ++kg) {
    if ((kg & 31) == 0) {                  // keep the weight stream ahead (L2)
      int pf = (kg + 32 < KG) ? kg + 32 : kg;
      __builtin_prefetch(&wmain[(size_t)pf * 64 + lane * 2], 0, 1);
    }
    v2f a  = *(const v2f*)&As[mrow * LDA + 4 * kg + koff];
    v2f bf = *(const v2f*)&wmain[(size_t)kg * 64 + lane * 2];
    acc = __builtin_amdgcn_wmma_f32_16x16x4_f32(
        false, a, false, bf, (short)0, acc, false, false);
  }
  int o = (lane & 15) + 16 * wv;
  #pragma unroll
  for (int r = 0; r < 8; ++r) {
    int m = r + ((lane < 16) ? 0 : 8);
    out[(size_t)(b * OUTC + o) * HWSZ + (hw + m)] = acc[r];
  }
}

// ---------------------------------------------------------------------------
extern "C" void kernel_launch(void* const* d_in, const int* in_sizes, int n_in,
                              void* d_out, int out_size, void* d_ws, size_t ws_size,
                              hipStream_t stream) {
  const float* x      = (const float*)d_in[0];
  const float* w_p    = (const float*)d_in[1];
  const float* b_p    = (const float*)d_in[2];
  const float* w_m    = (const float*)d_in[3];
  const float* b_m    = (const float*)d_in[4];
  const float* w_conv = (const float*)d_in[5];
  float* ws  = (float*)d_ws;               // needs ~7.6 MB
  float* out = (float*)d_out;
  float* offmod = ws + OFFMOD_OFF;

  const int packN = 2 * KG * 64 + 4 * KG * 64;          // 55296 elements
  pack_weights<<<(packN + 255) / 256, 256, 0, stream>>>(w_p, w_m, w_conv, ws);
  offmod_conv<<<BHW / 16, 128, 0, stream>>>(x, b_p, b_m, ws, offmod);
  deform_gemm<<<BHW / 16, 128, 0, stream>>>(x, ws, offmod, out);
}